// MoELayer_81209241632908
// MI455X (gfx1250) — compile-verified
//
#include <hip/hip_runtime.h>
#include <cstddef>

typedef __attribute__((ext_vector_type(2))) float v2f;
typedef __attribute__((ext_vector_type(8))) float v8f;

#define DEV __device__ __forceinline__

DEV float siluf(float x) { return x / (1.f + expf(-x)); }
DEV float geluf(float x) {
  float x3 = x * x * x;
  return 0.5f * x * (1.f + tanhf(0.7978845608028654f * (x + 0.044715f * x3)));
}
DEV float softplusf(float x) { return (x > 20.f) ? x : log1pf(expf(x)); }
DEV float apply_act(float v, int act) {
  if (act == 1) return geluf(v);
  if (act == 2) return softplusf(v);
  return v;
}

DEV v8f wmma_step(v2f a, v2f b, v8f acc) {
  return __builtin_amdgcn_wmma_f32_16x16x4_f32(false, a, false, b, (short)0,
                                               acc, false, false);
}

// Fully guarded 16x16 tile accumulation (edge tiles only).
DEV v8f wmma_tile_guarded(const float* __restrict__ A, int lda,
                          const float* __restrict__ W, int M, int N, int K,
                          int tm, int tnn, int lane) {
  const int mr = lane & 15;
  const int kb = (lane >> 4) << 1;
  const int m = tm + mr;
  const int n = tnn + mr;
  v8f acc = {};
  for (int k0 = 0; k0 < K; k0 += 4) {
    const int ka = k0 + kb;
    const int kc = ka + 1;
    const bool mok = (m < M);
    const bool nok = (n < N);
    v2f a, b;
    a.x = (mok && ka < K) ? A[(size_t)m * lda + ka] : 0.f;
    a.y = (mok && kc < K) ? A[(size_t)m * lda + kc] : 0.f;
    b.x = (nok && ka < K) ? W[(size_t)ka * N + n] : 0.f;
    b.y = (nok && kc < K) ? W[(size_t)kc * N + n] : 0.f;
    acc = wmma_step(a, b, acc);
  }
  return acc;
}

// ---------------------------------------------------------------------------
// C = act(A @ W + bias)
// A: M x K row-major (leading dim lda); W: K x N row-major; C: M x N (ldc).
// One wave32 per 16(M) x 32(N) tile: two accumulators share each A fragment.
// Fast path (tile fully in range, K multiple of 4 handled by tail) has zero
// bounds checks -> no EXEC save/restore in the inner loop, b64 A loads.
// act: 0 = none, 1 = gelu(tanh approx), 2 = softplus
// ---------------------------------------------------------------------------
__global__ void k_gemm32(const float* __restrict__ A, int lda,
                         const float* __restrict__ W,
                         const float* __restrict__ bias,
                         float* __restrict__ C, int ldc,
                         int M, int N, int K, int act) {
  const int lane = threadIdx.x & 31;
  const int tm = blockIdx.x << 4;
  const int tn = blockIdx.y << 5;
  const int mr = lane & 15;
  const int kb = (lane >> 4) << 1;  // lanes 0-15 -> K {0,1}; lanes 16-31 -> K {2,3}
  const int K4 = K & ~3;
  const int rbase = tm + ((lane >> 4) << 3);  // C rows p / p+8 across lane halves

  if ((tm + 16 <= M) && (tn + 32 <= N)) {
    // ---- fast path: no guards in the K loop ----
    const float* Ap = A + (size_t)(tm + mr) * lda + kb;
    const float* Wp0 = W + (size_t)kb * N + tn + mr;
    const float* Wp1 = Wp0 + 16;
    v8f acc0 = {}, acc1 = {};
    for (int k0 = 0; k0 < K4; k0 += 4) {
      v2f a = *(const v2f*)(Ap + k0);  // global_load_b64 (lda, kb even -> aligned)
      v2f b0, b1;
      b0.x = Wp0[(size_t)k0 * N];
      b0.y = Wp0[(size_t)(k0 + 1) * N];
      b1.x = Wp1[(size_t)k0 * N];
      b1.y = Wp1[(size_t)(k0 + 1) * N];
      acc0 = wmma_step(a, b0, acc0);
      acc1 = wmma_step(a, b1, acc1);
    }
    if (K4 < K) {  // ragged-K tail (only r=6 hits this)
      const int ka = K4 + kb;
      const int kc = ka + 1;
      v2f a, b0, b1;
      a.x = (ka < K) ? Ap[K4] : 0.f;
      a.y = (kc < K) ? Ap[K4 + 1] : 0.f;
      b0.x = (ka < K) ? Wp0[(size_t)K4 * N] : 0.f;
      b0.y = (kc < K) ? Wp0[(size_t)(K4 + 1) * N] : 0.f;
      b1.x = (ka < K) ? Wp1[(size_t)K4 * N] : 0.f;
      b1.y = (kc < K) ? Wp1[(size_t)(K4 + 1) * N] : 0.f;
      acc0 = wmma_step(a, b0, acc0);
      acc1 = wmma_step(a, b1, acc1);
    }
    const int col0 = tn + mr;
    const float bv0 = bias ? bias[col0] : 0.f;
    const float bv1 = bias ? bias[col0 + 16] : 0.f;
#pragma unroll
    for (int p = 0; p < 8; ++p) {
      const int row = rbase + p;
      C[(size_t)row * ldc + col0] = apply_act(acc0[p] + bv0, act);
      C[(size_t)row * ldc + col0 + 16] = apply_act(acc1[p] + bv1, act);
    }
  } else {
    // ---- edge path: fully guarded, two independent 16x16 subtiles ----
    for (int nt = 0; nt < 2; ++nt) {
      const int tnn = tn + (nt << 4);
      if (tnn >= N) continue;
      v8f acc = wmma_tile_guarded(A, lda, W, M, N, K, tm, tnn, lane);
      const int col = tnn + mr;
      if (col < N) {
        const float bv = bias ? bias[col] : 0.f;
#pragma unroll
        for (int p = 0; p < 8; ++p) {
          const int row = rbase + p;
          if (row < M) C[(size_t)row * ldc + col] = apply_act(acc[p] + bv, act);
        }
      }
    }
  }
}

// xc[b,t,d] = silu(conv_b[d] + xi[b,t-1,d]*w[d,0] + xi[b,t,d]*w[d,1])
// xi = first D columns of xiz (leading dim twoD)
__global__ void k_conv_silu(const float* __restrict__ xiz, int twoD,
                            const float* __restrict__ cw,
                            const float* __restrict__ cb,
                            float* __restrict__ xc, int Bn, int L, int D) {
  int i = blockIdx.x * blockDim.x + threadIdx.x;
  int total = Bn * L * D;
  if (i >= total) return;
  int d = i % D;
  int t = (i / D) % L;
  size_t row = (size_t)(i / D);
  float prev = (t > 0) ? xiz[(row - 1) * twoD + d] : 0.f;
  float cur = xiz[row * twoD + d];
  float v = cb[d] + prev * cw[d * 2 + 0] + cur * cw[d * 2 + 1];
  xc[row * D + d] = siluf(v);
}

// Sequential selective-state scan: one wave32 per (b,d); lane n holds h[n],
// n = 0..31 (D_STATE = 32 matches wave32 exactly).
// yg[b,t,d] = (sum_n h_t[n]*C_t[n] + D[d]*xc) * silu(z)
__global__ void k_scan(const float* __restrict__ dt, const float* __restrict__ xc,
                       const float* __restrict__ dbc, int ldd, int r,
                       const float* __restrict__ A_log, const float* __restrict__ Dp,
                       const float* __restrict__ xiz, int twoD, int D,
                       float* __restrict__ yg, int Bn, int L) {
  int wid = (blockIdx.x * blockDim.x + threadIdx.x) >> 5;
  int lane = threadIdx.x & 31;
  if (wid >= Bn * D) return;
  int b = wid / D;
  int d = wid % D;
  const float An = -expf(A_log[d * 32 + lane]);
  const float Dd = Dp[d];
  float h = 0.f;
  const size_t base = (size_t)b * L;
  for (int t = 0; t < L; ++t) {
    const size_t row = base + t;
    const float dtv = dt[row * D + d];
    const float xcv = xc[row * D + d];
    const float Bv = dbc[row * ldd + r + lane];
    const float Cv = dbc[row * ldd + r + 32 + lane];
    h = expf(dtv * An) * h + (dtv * xcv) * Bv;
    float v = h * Cv;
#pragma unroll
    for (int off = 16; off > 0; off >>= 1) v += __shfl_xor(v, off, 32);
    if (lane == 0) {
      const float z = xiz[row * twoD + D + d];
      yg[row * D + d] = (v + Dd * xcv) * siluf(z);
    }
  }
}

// out[b, L-1-t, :] = in[b, t, :]
__global__ void k_flip(const float* __restrict__ in, float* __restrict__ out,
                       int Bn, int L, int D) {
  int i = blockIdx.x * blockDim.x + threadIdx.x;
  int total = Bn * L * D;
  if (i >= total) return;
  int d = i % D;
  int t = (i / D) % L;
  int b = i / (D * L);
  out[((size_t)(b * L + (L - 1 - t))) * D + d] = in[i];
}

// out = LN(x [+ a] [+ b]) row-wise over D; one wave32 per row. In-place safe.
__global__ void k_add_ln(const float* __restrict__ x, const float* __restrict__ a,
                         const float* __restrict__ b, const float* __restrict__ g,
                         const float* __restrict__ beta, float* __restrict__ out,
                         int rows, int D) {
  int wid = (blockIdx.x * blockDim.x + threadIdx.x) >> 5;
  int lane = threadIdx.x & 31;
  if (wid >= rows) return;
  const size_t base = (size_t)wid * D;
  float s = 0.f, ss = 0.f;
  for (int d = lane; d < D; d += 32) {
    float v = x[base + d];
    if (a) v += a[base + d];
    if (b) v += b[base + d];
    s += v;
    ss += v * v;
  }
#pragma unroll
  for (int off = 16; off > 0; off >>= 1) {
    s += __shfl_xor(s, off, 32);
    ss += __shfl_xor(ss, off, 32);
  }
  const float mu = s / (float)D;
  const float var = ss / (float)D - mu * mu;
  const float rstd = rsqrtf(var + 1e-5f);
  for (int d = lane; d < D; d += 32) {
    float v = x[base + d];
    if (a) v += a[base + d];
    if (b) v += b[base + d];
    out[base + d] = (v - mu) * rstd * g[d] + beta[d];
  }
}

// in: (B, R, C) -> out: (B, C, R)
__global__ void k_transpose(const float* __restrict__ in, float* __restrict__ out,
                            int Bn, int R, int C) {
  int i = blockIdx.x * blockDim.x + threadIdx.x;
  int total = Bn * R * C;
  if (i >= total) return;
  int c = i % C;
  int rr = (i / C) % R;
  int b = i / (C * R);
  out[((size_t)b * C + c) * R + rr] = in[i];
}

// pooled[b,t] = mean_j x[b,j,t]; logits = pooled @ gw^T + gb; sel[b] = argmax
__global__ void k_gate(const float* __restrict__ x, const float* __restrict__ gw,
                       const float* __restrict__ gb, int* __restrict__ sel,
                       int Bn, int J, int T, int E) {
  __shared__ float red[4][256];
  int b = blockIdx.x;
  float part[4] = {0.f, 0.f, 0.f, 0.f};
  for (int t = threadIdx.x; t < T; t += blockDim.x) {
    float p = 0.f;
    for (int j = 0; j < J; ++j) p += x[((size_t)b * J + j) * T + t];
    p /= (float)J;
    for (int e = 0; e < E; ++e) part[e] += p * gw[e * T + t];
  }
  for (int e = 0; e < E; ++e) red[e][threadIdx.x] = part[e];
  __syncthreads();
  if (threadIdx.x == 0) {
    int best = 0;
    float bv = -1e30f;
    for (int e = 0; e < E; ++e) {
      float s = gb[e];
      for (int i2 = 0; i2 < 256; ++i2) s += red[e][i2];
      if (s > bv) { bv = s; best = e; }  // strict > = first-of-ties like top_k
    }
    sel[b] = best;
  }
}

// top-1 softmax weight == 1.0, so output = selected expert verbatim
__global__ void k_select(const float* __restrict__ e0, const float* __restrict__ e1,
                         const float* __restrict__ e2, const float* __restrict__ e3,
                         const int* __restrict__ sel, float* __restrict__ out,
                         int Bn, int per) {
  int i = blockIdx.x * blockDim.x + threadIdx.x;
  if (i >= Bn * per) return;
  int b = i / per;
  int s = sel[b];
  const float* src = (s == 0) ? e0 : (s == 1) ? e1 : (s == 2) ? e2 : e3;
  out[i] = src[i];
}

// ---------------------------------------------------------------------------
// Host-side orchestration
// ---------------------------------------------------------------------------
struct MambaP {
  const float *A_log, *D, *W_dt, *W_in, *W_out, *W_x, *b_dt, *conv_b, *conv_w;
};
struct LayerP {
  const float *c1_b, *c1_w, *c2_b, *c2_w;
  MambaP m1, m2;
  const float *n1_b, *n1_g, *n2_b, *n2_g;
};
struct EncP {
  LayerP L[3];
  const float *n_b, *n_g;
};

static MambaP parse_mamba(void* const* d_in, int& i) {
  MambaP m;  // jax dict sorted-key order
  m.A_log = (const float*)d_in[i++];
  m.D = (const float*)d_in[i++];
  m.W_dt = (const float*)d_in[i++];
  m.W_in = (const float*)d_in[i++];
  m.W_out = (const float*)d_in[i++];
  m.W_x = (const float*)d_in[i++];
  m.b_dt = (const float*)d_in[i++];
  m.conv_b = (const float*)d_in[i++];
  m.conv_w = (const float*)d_in[i++];
  return m;
}
static LayerP parse_layer(void* const* d_in, int& i) {
  LayerP l;
  l.c1_b = (const float*)d_in[i++];
  l.c1_w = (const float*)d_in[i++];
  l.c2_b = (const float*)d_in[i++];
  l.c2_w = (const float*)d_in[i++];
  l.m1 = parse_mamba(d_in, i);
  l.m2 = parse_mamba(d_in, i);
  l.n1_b = (const float*)d_in[i++];
  l.n1_g = (const float*)d_in[i++];
  l.n2_b = (const float*)d_in[i++];
  l.n2_g = (const float*)d_in[i++];
  return l;
}
static EncP parse_enc(void* const* d_in, int& i) {
  EncP e;  // keys sorted: 'layers', 'n_b', 'n_g'
  for (int l = 0; l < 3; ++l) e.L[l] = parse_layer(d_in, i);
  e.n_b = (const float*)d_in[i++];
  e.n_g = (const float*)d_in[i++];
  return e;
}

struct Scratch {
  float *xiz, *xc, *dbc, *dt, *yg;
  float *m1o, *m2o, *fbuf, *fbuf2;
  float *hbuf, *ybuf, *tmpT;
};

static inline int cdiv(int a, int b) { return (a + b - 1) / b; }

static void launch_gemm(const float* A, int lda, const float* W,
                        const float* bias, float* C, int ldc, int M, int N,
                        int K, int act, hipStream_t st) {
  k_gemm32<<<dim3(cdiv(M, 16), cdiv(N, 32)), 32, 0, st>>>(A, lda, W, bias, C,
                                                          ldc, M, N, K, act);
}

static void run_mamba(const MambaP& mp, const float* xin, float* yout, int B,
                      int L, int D, int r, Scratch& sc, hipStream_t st) {
  const int M = B * L, twoD = 2 * D, ldd = r + 64;
  launch_gemm(xin, D, mp.W_in, nullptr, sc.xiz, twoD, M, twoD, D, 0, st);
  const int tot = M * D;
  k_conv_silu<<<cdiv(tot, 256), 256, 0, st>>>(sc.xiz, twoD, mp.conv_w, mp.conv_b,
                                              sc.xc, B, L, D);
  launch_gemm(sc.xc, D, mp.W_x, nullptr, sc.dbc, ldd, M, ldd, D, 0, st);
  launch_gemm(sc.dbc, ldd, mp.W_dt, mp.b_dt, sc.dt, D, M, D, r, 2, st);
  const int waves = B * D;
  k_scan<<<cdiv(waves * 32, 256), 256, 0, st>>>(sc.dt, sc.xc, sc.dbc, ldd, r,
                                                mp.A_log, mp.D, sc.xiz, twoD, D,
                                                sc.yg, B, L);
  launch_gemm(sc.yg, D, mp.W_out, nullptr, yout, D, M, D, D, 0, st);
}

static void run_layer(const LayerP& lp, float* x, int B, int L, int D, int r,
                      int dff, Scratch& sc, hipStream_t st) {
  const int rows = B * L, tot = rows * D;
  run_mamba(lp.m1, x, sc.m1o, B, L, D, r, sc, st);
  k_flip<<<cdiv(tot, 256), 256, 0, st>>>(x, sc.fbuf, B, L, D);
  run_mamba(lp.m2, sc.fbuf, sc.fbuf2, B, L, D, r, sc, st);
  k_flip<<<cdiv(tot, 256), 256, 0, st>>>(sc.fbuf2, sc.m2o, B, L, D);
  k_add_ln<<<cdiv(rows * 32, 256), 256, 0, st>>>(x, sc.m1o, sc.m2o, lp.n1_g,
                                                 lp.n1_b, x, rows, D);
  launch_gemm(x, D, lp.c1_w, lp.c1_b, sc.hbuf, dff, rows, dff, D, 1, st);
  launch_gemm(sc.hbuf, dff, lp.c2_w, lp.c2_b, sc.ybuf, D, rows, D, dff, 0, st);
  k_add_ln<<<cdiv(rows * 32, 256), 256, 0, st>>>(x, sc.ybuf, nullptr, lp.n2_g,
                                                 lp.n2_b, x, rows, D);
}

static void run_encoder(const EncP& ep, float* x, int B, int L, int D, int r,
                        int dff, Scratch& sc, hipStream_t st) {
  for (int l = 0; l < 3; ++l) run_layer(ep.L[l], x, B, L, D, r, dff, sc, st);
  const int rows = B * L;
  k_add_ln<<<cdiv(rows * 32, 256), 256, 0, st>>>(x, nullptr, nullptr, ep.n_g,
                                                 ep.n_b, x, rows, D);
}

extern "C" void kernel_launch(void* const* d_in, const int* in_sizes, int n_in,
                              void* d_out, int out_size, void* d_ws,
                              size_t ws_size, hipStream_t stream) {
  (void)in_sizes; (void)n_in; (void)out_size; (void)ws_size;
  const int B = 4, J = 96, T = 192;
  const size_t NE = (size_t)B * J * T;  // 73728

  int idx = 0;
  const float* x = (const float*)d_in[idx++];
  EncP tp = parse_enc(d_in, idx);                 // d=96,  r=6,  dff=384, seq=192
  EncP sp = parse_enc(d_in, idx);                 // d=192, r=12, dff=768, seq=96
  const float* gate_w = (const float*)d_in[idx++];
  const float* gate_b = (const float*)d_in[idx++];

  char* wsp = (char*)d_ws;
  auto allocf = [&](size_t n) -> float* {
    float* p = (float*)wsp;
    wsp += ((n * sizeof(float) + 255) / 256) * 256;
    return p;
  };

  Scratch sc;
  sc.xiz = allocf(768 * 384);   // max rows*2D = 147456 (both configs)
  sc.xc = allocf(NE);
  sc.dbc = allocf(768 * 80);    // max rows*(r+64)
  sc.dt = allocf(NE);
  sc.yg = allocf(NE);
  sc.m1o = allocf(NE);
  sc.m2o = allocf(NE);
  sc.fbuf = allocf(NE);
  sc.fbuf2 = allocf(NE);
  sc.hbuf = allocf(768 * 384);  // max rows*dff = 294912
  sc.ybuf = allocf(NE);
  sc.tmpT = allocf(NE);
  float* t1 = allocf(NE);
  float* s1 = allocf(NE);
  float* ts = allocf(NE);
  float* st_ = allocf(NE);
  float* ss = allocf(NE);
  float* tt = allocf(NE);
  int* sel = (int*)allocf(64);

  const int tot = (int)NE;

  auto temporal_apply = [&](const float* in, float* out) {
    k_transpose<<<cdiv(tot, 256), 256, 0, stream>>>(in, sc.tmpT, B, J, T);
    run_encoder(tp, sc.tmpT, B, /*L*/ T, /*D*/ J, /*r*/ 6, /*dff*/ 4 * J, sc, stream);
    k_transpose<<<cdiv(tot, 256), 256, 0, stream>>>(sc.tmpT, out, B, T, J);
  };
  auto spatial_apply = [&](const float* in, float* out) {
    hipMemcpyAsync(out, in, NE * sizeof(float), hipMemcpyDeviceToDevice, stream);
    run_encoder(sp, out, B, /*L*/ J, /*D*/ T, /*r*/ 12, /*dff*/ 4 * T, sc, stream);
  };

  temporal_apply(x, t1);
  spatial_apply(x, s1);
  spatial_apply(t1, ts);
  temporal_apply(t1, tt);
  temporal_apply(s1, st_);
  spatial_apply(s1, ss);

  k_gate<<<B, 256, 0, stream>>>(x, gate_w, gate_b, sel, B, J, T, 4);
  k_select<<<cdiv(tot, 256), 256, 0, stream>>>(ts, st_, ss, tt, sel,
                                               (float*)d_out, B, J * T);
}